// HPNF_7146825580852
// MI455X (gfx1250) — compile-verified
//
#include <hip/hip_runtime.h>

typedef float v2f __attribute__((ext_vector_type(2)));
typedef float v8f __attribute__((ext_vector_type(8)));

#define NN 100000
#define CH 128
#define NG 128
#define BN_EPS 1e-5f
#define LPAD 132                     // LDS row stride (floats), breaks bank conflicts
#define GEMM_LDS ((128 * LPAD + 8 * 16 * LPAD) * 4)  // W tile + 8 wave A-strips

// ---------------- utility kernels ----------------

__global__ void k_fill(float* __restrict__ p, float v, int n) {
    int i = blockIdx.x * blockDim.x + threadIdx.x;
    if (i < n) p[i] = v;
}

__global__ void k_edge_deg(const int* __restrict__ dst, float* __restrict__ deg, int ne) {
    int e = blockIdx.x * blockDim.x + threadIdx.x;
    if (e < ne) atomicAdd(&deg[dst[e]], 1.0f);
}

__global__ void k_dinv(const float* __restrict__ deg, float* __restrict__ dinv, int n) {
    int i = blockIdx.x * blockDim.x + threadIdx.x;
    if (i < n) {
        float d = deg[i];
        dinv[i] = d > 0.0f ? rsqrtf(d) : 0.0f;
    }
}

// ---------------- fp32 WMMA GEMM: out[M x 128] = A[M x 128] @ W[128 x 128] ----------------
// 256 threads = 8 wave32; each wave owns one 16-row strip, 8 accumulator tiles of 16x16.
__global__ void k_gemm128(const float* __restrict__ A, const float* __restrict__ W,
                          float* __restrict__ out, int nstrips) {
    extern __shared__ float lds[];
    float* sW = lds;                 // 128 x LPAD
    float* sA = lds + 128 * LPAD;    // 8 waves x 16 x LPAD

    const int tid  = threadIdx.x;
    const int lane = tid & 31;
    const int wave = tid >> 5;

    // stage W (row-major [k][n]) into LDS, padded
    for (int e = tid; e < 128 * 128; e += 256) {
        int k = e >> 7, n = e & 127;
        sW[k * LPAD + n] = W[e];
    }

    const int strip  = blockIdx.x * 8 + wave;
    const bool active = strip < nstrips;
    float* sAw = sA + wave * (16 * LPAD);
    if (active) {
        const float* Ab = A + (size_t)strip * 16 * 128;
        for (int i = 0; i < 64; ++i) {           // 2048 elems, coalesced
            int e = i * 32 + lane;
            sAw[(e >> 7) * LPAD + (e & 127)] = Ab[e];
        }
    }
    __syncthreads();
    if (!active) return;

    const int lh = lane >> 4;   // half-wave select (K pairs)
    const int lr = lane & 15;   // row / col within tile

    v8f acc[8] = {};
    for (int k0 = 0; k0 < 128; k0 += 4) {
        const int ka = k0 + lh * 2;
        v2f a;
        a.x = sAw[lr * LPAD + ka];
        a.y = sAw[lr * LPAD + ka + 1];
#pragma unroll
        for (int j = 0; j < 8; ++j) {
            v2f b;
            b.x = sW[ka * LPAD + j * 16 + lr];
            b.y = sW[(ka + 1) * LPAD + j * 16 + lr];
            acc[j] = __builtin_amdgcn_wmma_f32_16x16x4_f32(
                false, a, false, b, (short)0, acc[j], false, false);
        }
    }

    float* ob = out + (size_t)strip * 16 * 128;
#pragma unroll
    for (int j = 0; j < 8; ++j) {
#pragma unroll
        for (int v = 0; v < 8; ++v) {
            int r = v + 8 * lh;          // C layout: VGPR v -> M=v (lanes 0-15) / v+8 (16-31)
            int c = j * 16 + lr;
            ob[r * 128 + c] = acc[j][v];
        }
    }
}

// out[i][c] = dinv[i]^2 * xw[i][c]   (self-loop term; also zero-initializes scatter target)
__global__ void k_selfloop(const float* __restrict__ xw, const float* __restrict__ dinv,
                           float* __restrict__ out, int total) {
    int i = blockIdx.x * blockDim.x + threadIdx.x;
    if (i < total) {
        float dv = dinv[i >> 7];
        out[i] = dv * dv * xw[i];
    }
}

// one wave per edge: out[dst] += dinv[s]*dinv[d] * xw[src]  (float4 per lane)
__global__ void k_prop(const float* __restrict__ xw, const float* __restrict__ dinv,
                       const int* __restrict__ src, const int* __restrict__ dst,
                       float* __restrict__ out, int ne) {
    int gid  = blockIdx.x * blockDim.x + threadIdx.x;
    int e    = gid >> 5;
    int lane = gid & 31;
    if (e >= ne) return;
    int s = src[e], d = dst[e];
    float c = dinv[s] * dinv[d];
    float4 v = ((const float4*)(xw + (size_t)s * CH))[lane];
    float* o = out + (size_t)d * CH + lane * 4;
    atomicAdd(o + 0, v.x * c);
    atomicAdd(o + 1, v.y * c);
    atomicAdd(o + 2, v.z * c);
    atomicAdd(o + 3, v.w * c);
}

// h = relu((h + b - rmean) * rsqrt(rvar+eps) * gamma + beta), in place
__global__ void k_bnrelu(float* __restrict__ h, const float* __restrict__ b,
                         const float* __restrict__ gamma, const float* __restrict__ beta,
                         const float* __restrict__ rmean, const float* __restrict__ rvar,
                         int total) {
    int i = blockIdx.x * blockDim.x + threadIdx.x;
    if (i < total) {
        int c = i & (CH - 1);
        float v = (h[i] + b[c] - rmean[c]) * rsqrtf(rvar[c] + BN_EPS) * gamma[c] + beta[c];
        h[i] = fmaxf(v, 0.0f);
    }
}

// one wave per node: sums[batch[i]] += h[i] + b2 ; cnts[batch[i]] += 1
__global__ void k_pool(const float* __restrict__ h, const float* __restrict__ b2,
                       const int* __restrict__ batch, float* __restrict__ sums,
                       float* __restrict__ cnts, int nn) {
    int gid  = blockIdx.x * blockDim.x + threadIdx.x;
    int node = gid >> 5;
    int lane = gid & 31;
    if (node >= nn) return;
    int g = batch[node];
    float4 v = ((const float4*)(h + (size_t)node * CH))[lane];
    const float4 bb = ((const float4*)b2)[lane];
    float* o = sums + (size_t)g * CH + lane * 4;
    atomicAdd(o + 0, v.x + bb.x);
    atomicAdd(o + 1, v.y + bb.y);
    atomicAdd(o + 2, v.z + bb.z);
    atomicAdd(o + 3, v.w + bb.w);
    if (lane == 0) atomicAdd(&cnts[g], 1.0f);
}

// per-graph head: pooled = sums/max(cnt,1); z = relu(pooled @ fw1 + fb1); out = z @ cw + cb
__global__ void k_mlp(const float* __restrict__ sums, const float* __restrict__ cnts,
                      const float* __restrict__ fw1, const float* __restrict__ fb1,
                      const float* __restrict__ cw, const float* __restrict__ cb,
                      float* __restrict__ out) {
    __shared__ float p[CH];
    __shared__ float z[64];
    int g = blockIdx.x, tid = threadIdx.x;
    float inv = 1.0f / fmaxf(cnts[g], 1.0f);
    for (int c = tid; c < CH; c += 64) p[c] = sums[g * CH + c] * inv;
    __syncthreads();
    float acc = fb1[tid];
    for (int k = 0; k < CH; ++k) acc += p[k] * fw1[k * 64 + tid];
    z[tid] = fmaxf(acc, 0.0f);
    __syncthreads();
    if (tid < 2) {
        float o = cb[tid];
        for (int j = 0; j < 64; ++j) o += z[j] * cw[j * 2 + tid];
        out[g * 2 + tid] = o;
    }
}

// ---------------- launch ----------------

extern "C" void kernel_launch(void* const* d_in, const int* in_sizes, int n_in,
                              void* d_out, int out_size, void* d_ws, size_t ws_size,
                              hipStream_t stream) {
    const float* x     = (const float*)d_in[0];
    const int*   ei    = (const int*)d_in[1];
    const int*   batch = (const int*)d_in[2];
    const float* W1    = (const float*)d_in[3];
    const float* b1    = (const float*)d_in[4];
    const float* gamma = (const float*)d_in[5];
    const float* beta  = (const float*)d_in[6];
    const float* rmean = (const float*)d_in[7];
    const float* rvar  = (const float*)d_in[8];
    const float* W2    = (const float*)d_in[9];
    const float* b2    = (const float*)d_in[10];
    const float* fw1   = (const float*)d_in[11];
    const float* fb1   = (const float*)d_in[12];
    const float* cw    = (const float*)d_in[13];
    const float* cb    = (const float*)d_in[14];
    float* out = (float*)d_out;

    const int ne = in_sizes[1] / 2;
    const int* src = ei;
    const int* dst = ei + ne;

    float* ws   = (float*)d_ws;
    float* bufA = ws;                        // NN*CH : GEMM outputs (xw / xw2)
    float* bufB = bufA + (size_t)NN * CH;    // NN*CH : scatter target / h1
    float* deg  = bufB + (size_t)NN * CH;    // NN
    float* dinv = deg + NN;                  // NN
    float* sums = dinv + NN;                 // NG*CH
    float* cnts = sums + (size_t)NG * CH;    // NG  (contiguous with sums)

    (void)hipFuncSetAttribute((const void*)k_gemm128,
                              hipFuncAttributeMaxDynamicSharedMemorySize, GEMM_LDS);

    const int T = 256;
    const int nstrips = NN / 16;             // 6250
    const int gemmBlocks = (nstrips + 7) / 8;

    // degrees (self-loop => init 1) and normalization
    k_fill<<<(NN + T - 1) / T, T, 0, stream>>>(deg, 1.0f, NN);
    k_fill<<<(NG * CH + NG + T - 1) / T, T, 0, stream>>>(sums, 0.0f, NG * CH + NG);
    k_edge_deg<<<(ne + T - 1) / T, T, 0, stream>>>(dst, deg, ne);
    k_dinv<<<(NN + T - 1) / T, T, 0, stream>>>(deg, dinv, NN);

    // GCNConv 1
    k_gemm128<<<gemmBlocks, T, GEMM_LDS, stream>>>(x, W1, bufA, nstrips);
    k_selfloop<<<(NN * CH + T - 1) / T, T, 0, stream>>>(bufA, dinv, bufB, NN * CH);
    k_prop<<<(ne * 32 + T - 1) / T, T, 0, stream>>>(bufA, dinv, src, dst, bufB, ne);
    k_bnrelu<<<(NN * CH + T - 1) / T, T, 0, stream>>>(bufB, b1, gamma, beta, rmean, rvar, NN * CH);

    // GCNConv 2
    k_gemm128<<<gemmBlocks, T, GEMM_LDS, stream>>>(bufB, W2, bufA, nstrips);
    k_selfloop<<<(NN * CH + T - 1) / T, T, 0, stream>>>(bufA, dinv, bufB, NN * CH);
    k_prop<<<(ne * 32 + T - 1) / T, T, 0, stream>>>(bufA, dinv, src, dst, bufB, ne);

    // pool + head
    k_pool<<<((size_t)NN * 32 + T - 1) / T, T, 0, stream>>>(bufB, b2, batch, sums, cnts, NN);
    k_mlp<<<NG, 64, 0, stream>>>(sums, cnts, fw1, fb1, cw, cb, out);
}